// CudaMemory_34187939676717
// MI455X (gfx1250) — compile-verified
//
#include <hip/hip_runtime.h>
#include <hip/hip_bf16.h>
#include <math.h>

// ---------------- problem constants ----------------
#define B_   128
#define I_   512
#define N_   512
#define W_   32
#define R_   4
#define C_   256          // padded interface columns
#define NCOLS 247         // real interface columns
#define EPS_ 1e-6f

// d_out float offsets (return order: rv, memory, link, precedence, read_weights, ww, usage)
#define OUT_RV    0
#define OUT_MEM   (OUT_RV   + B_*R_*W_)      // 16384
#define OUT_LINK  (OUT_MEM  + B_*N_*W_)      // 2113536
#define OUT_PREC  (OUT_LINK + (size_t)B_*N_*N_)
#define OUT_RW    (OUT_PREC + B_*N_)
#define OUT_WW    (OUT_RW   + B_*R_*N_)
#define OUT_USAGE (OUT_WW   + B_*N_)

// workspace float offsets
#define WS_WCAT  0
#define WS_BCAT  (WS_WCAT + C_*I_)           // 131072
#define WS_IFC   (WS_BCAT + C_)              // +256
#define WS_ALLOC (WS_IFC  + B_*C_)           // +32768
#define WS_SWW   (WS_ALLOC+ B_*N_)           // +65536
#define WS_RSIM  (WS_SWW  + B_)              // +128
#define WS_FW    (WS_RSIM + B_*R_*N_)        // +262144
#define WS_BW    (WS_FW   + B_*R_*N_)

typedef __attribute__((ext_vector_type(2))) float v2f;
typedef __attribute__((ext_vector_type(8))) float v8f;

// ---------------- activations ----------------
__device__ __forceinline__ float sigm_(float x) { return 1.0f / (1.0f + __expf(-x)); }
__device__ __forceinline__ float sftp_(float x) { return (x > 20.0f) ? x : log1pf(expf(x)); }
__device__ __forceinline__ float head_act(int c, float x) {
  if (c < 128) return tanhf(x);   // read_keys
  if (c < 132) return sftp_(x);   // read_str
  if (c < 164) return tanhf(x);   // write_key
  if (c < 165) return sftp_(x);   // write_str
  if (c < 197) return sigm_(x);   // erase
  if (c < 229) return tanhf(x);   // write_vec
  if (c < 235) return sigm_(x);   // free gates, alloc gate, write gate
  return x;                       // read modes (softmax later)
}

// ---------------- K0: pack 10 head weight matrices into [256,512] ----------------
__global__ void pack_kernel(const float* Wrk, const float* Wrs, const float* Wwk,
                            const float* Wws, const float* Wev, const float* Wwv,
                            const float* Wfg, const float* Wag, const float* Wwg,
                            const float* Wrm,
                            const float* brk, const float* brs, const float* bwk,
                            const float* bws, const float* bev, const float* bwv,
                            const float* bfg, const float* bag, const float* bwg,
                            const float* brm,
                            float* Wcat, float* bcat) {
  int t = blockIdx.x * blockDim.x + threadIdx.x;
  if (t >= C_ * I_) return;
  int row = t / I_, k = t % I_;
  if (row >= NCOLS) { Wcat[row * I_ + k] = 0.0f; if (k == 0) bcat[row] = 0.0f; return; }
  const float *Wp, *bp; int off;
  if      (row < 128) { Wp = Wrk; bp = brk; off = 0;   }
  else if (row < 132) { Wp = Wrs; bp = brs; off = 128; }
  else if (row < 164) { Wp = Wwk; bp = bwk; off = 132; }
  else if (row < 165) { Wp = Wws; bp = bws; off = 164; }
  else if (row < 197) { Wp = Wev; bp = bev; off = 165; }
  else if (row < 229) { Wp = Wwv; bp = bwv; off = 197; }
  else if (row < 233) { Wp = Wfg; bp = bfg; off = 229; }
  else if (row < 234) { Wp = Wag; bp = bag; off = 233; }
  else if (row < 235) { Wp = Wwg; bp = bwg; off = 234; }
  else                { Wp = Wrm; bp = brm; off = 235; }
  int lr = row - off;
  Wcat[row * I_ + k] = Wp[lr * I_ + k];
  if (k == 0) bcat[row] = bp[lr];
}

// ---------------- K1: interface heads GEMM via V_WMMA_F32_16X16X4_F32 ----------------
// C[m=batch 128][n=col 256] = xi[128,512] x Wcat^T ; bias + per-column activation fused.
__global__ __launch_bounds__(32) void heads_gemm_kernel(const float* __restrict__ xi,
                                                        const float* __restrict__ Wcat,
                                                        const float* __restrict__ bcat,
                                                        float* __restrict__ ifc) {
  int m0 = blockIdx.x * 16;
  int n0 = blockIdx.y * 16;
  int lane = threadIdx.x;          // wave32: one wave per 16x16 C tile
  int half = lane >> 4;            // K-pair selector per ISA A/B layout
  int l16  = lane & 15;
  const float* arow = xi   + (size_t)(m0 + l16) * I_ + half * 2;
  const float* brow = Wcat + (size_t)(n0 + l16) * I_ + half * 2;  // B[k][n] = Wcat[n][k]
  v8f acc = {};
  for (int k = 0; k < I_; k += 4) {
    v2f a = *(const v2f*)(arow + k);   // K = k+{0,1} (lanes 0-15) / k+{2,3} (lanes 16-31)
    v2f b = *(const v2f*)(brow + k);
    acc = __builtin_amdgcn_wmma_f32_16x16x4_f32(
        /*neg_a=*/false, a, /*neg_b=*/false, b,
        /*c_mod=*/(short)0, acc, /*reuse_a=*/false, /*reuse_b=*/false);
  }
  int col = n0 + l16;
  if (col < NCOLS) {
    float bias = bcat[col];
    for (int j = 0; j < 8; ++j) {
      int m = m0 + j + half * 8;     // C layout: VGPR j -> M=j (lo half) / j+8 (hi half)
      ifc[m * C_ + col] = head_act(col, acc[j] + bias);
    }
  }
}

// ---------------- K2: usage update (memory retention psi) ----------------
__global__ void usage_kernel(const float* __restrict__ ifc, const float* __restrict__ rw_old,
                             const float* __restrict__ wwgt, const float* __restrict__ usage_old,
                             float* __restrict__ usage_out) {
  int t = blockIdx.x * blockDim.x + threadIdx.x;
  if (t >= B_ * N_) return;
  int b = t / N_, n = t % N_;
  float psi = 1.0f;
  for (int r = 0; r < R_; ++r) {
    float fr = ifc[b * C_ + 229 + r];
    psi *= 1.0f - fr * rw_old[(b * R_ + r) * N_ + n];
  }
  float u = usage_old[t], w = wwgt[t];
  usage_out[t] = (u + w - u * w) * psi;
}

// ---------------- K3: allocation weighting (bitonic sort + exclusive cumprod) ----------------
__global__ __launch_bounds__(512) void alloc_kernel(const float* __restrict__ usage,
                                                    float* __restrict__ alloc) {
  __shared__ float v[512];
  __shared__ int   id[512];
  __shared__ float pe[512];
  int b = blockIdx.x, t = threadIdx.x;
  v[t] = usage[b * N_ + t]; id[t] = t;
  __syncthreads();
  for (int k = 2; k <= 512; k <<= 1) {
    for (int j = k >> 1; j > 0; j >>= 1) {
      int l = t ^ j;
      if (l > t) {
        bool asc = ((t & k) == 0);
        float a0 = v[t], a1 = v[l];
        if (asc ? (a0 > a1) : (a0 < a1)) {
          v[t] = a1; v[l] = a0;
          int i0 = id[t]; id[t] = id[l]; id[l] = i0;
        }
      }
      __syncthreads();
    }
  }
  if (t == 0) { float p = 1.0f; for (int i = 0; i < 512; ++i) { pe[i] = p; p *= v[i]; } }
  __syncthreads();
  alloc[b * N_ + id[t]] = (1.0f - v[t]) * pe[t];
}

// ---------------- K4: write content softmax, ww, precedence ----------------
__global__ __launch_bounds__(512) void ww_kernel(const float* __restrict__ ifc,
                                                 const float* __restrict__ mem_old,
                                                 const float* __restrict__ alloc,
                                                 const float* __restrict__ prec_old,
                                                 float* __restrict__ ww_out,
                                                 float* __restrict__ prec_out,
                                                 float* __restrict__ sww) {
  __shared__ float wk[32];
  __shared__ float red[512];
  __shared__ float wknorm;
  int b = blockIdx.x, t = threadIdx.x;
  if (t < 32) wk[t] = ifc[b * C_ + 132 + t];
  __syncthreads();
  if (t == 0) { float s = 0.f; for (int w = 0; w < 32; ++w) s += wk[w] * wk[w]; wknorm = sqrtf(s); }
  __syncthreads();
  float wstr = ifc[b * C_ + 164];
  float ag   = ifc[b * C_ + 233];
  float wg   = ifc[b * C_ + 234];
  const float* mr = mem_old + ((size_t)b * N_ + t) * W_;
  float dot = 0.f, nn = 0.f;
  for (int w = 0; w < 32; ++w) { float mv = mr[w]; dot += mv * wk[w]; nn += mv * mv; }
  float sc = wstr * (dot / (sqrtf(nn) * wknorm + EPS_));
  red[t] = sc; __syncthreads();
  for (int s = 256; s > 0; s >>= 1) { if (t < s) red[t] = fmaxf(red[t], red[t + s]); __syncthreads(); }
  float mx = red[0]; __syncthreads();
  float ex = __expf(sc - mx);
  red[t] = ex; __syncthreads();
  for (int s = 256; s > 0; s >>= 1) { if (t < s) red[t] += red[t + s]; __syncthreads(); }
  float wc = ex / red[0]; __syncthreads();
  float wwv = wg * (ag * alloc[b * N_ + t] + (1.0f - ag) * wc);
  ww_out[b * N_ + t] = wwv;
  red[t] = wwv; __syncthreads();
  for (int s = 256; s > 0; s >>= 1) { if (t < s) red[t] += red[t + s]; __syncthreads(); }
  float sw = red[0];
  if (t == 0) sww[b] = sw;
  prec_out[b * N_ + t] = (1.0f - sw) * prec_old[b * N_ + t] + wwv;
}

// ---------------- K5: memory erase/add + read-content similarity (fused) ----------------
__global__ __launch_bounds__(256) void memwrite_kernel(const float* __restrict__ ifc,
                                                       const float* __restrict__ mem_in,
                                                       const float* __restrict__ ww,
                                                       float* __restrict__ mem_out,
                                                       float* __restrict__ rsim) {
  __shared__ float er[32], wv[32], rk[128], rkn[4];
  int b = blockIdx.y, t = threadIdx.x;
  int n = blockIdx.x * 256 + t;
  if (t < 32) { er[t] = ifc[b * C_ + 165 + t]; wv[t] = ifc[b * C_ + 197 + t]; }
  if (t < 128) rk[t] = ifc[b * C_ + t];
  __syncthreads();
  if (t < 4) { float s = 0.f; for (int w = 0; w < 32; ++w) { float v = rk[t * 32 + w]; s += v * v; } rkn[t] = sqrtf(s); }
  __syncthreads();
  float wwv = ww[b * N_ + n];
  const float4* mi = (const float4*)(mem_in + ((size_t)b * N_ + n) * W_);
  float4*       mo = (float4*)(mem_out + ((size_t)b * N_ + n) * W_);
  float nn = 0.f, d0 = 0.f, d1 = 0.f, d2 = 0.f, d3 = 0.f;
  for (int q = 0; q < W_ / 4; ++q) {
    float4 mv = mi[q];
    float4 o;
    int w = q * 4;
    o.x = mv.x * (1.0f - wwv * er[w + 0]) + wwv * wv[w + 0];
    o.y = mv.y * (1.0f - wwv * er[w + 1]) + wwv * wv[w + 1];
    o.z = mv.z * (1.0f - wwv * er[w + 2]) + wwv * wv[w + 2];
    o.w = mv.w * (1.0f - wwv * er[w + 3]) + wwv * wv[w + 3];
    mo[q] = o;
    nn += o.x * o.x + o.y * o.y + o.z * o.z + o.w * o.w;
    d0 += o.x * rk[w] + o.y * rk[w + 1] + o.z * rk[w + 2] + o.w * rk[w + 3];
    d1 += o.x * rk[32 + w] + o.y * rk[33 + w] + o.z * rk[34 + w] + o.w * rk[35 + w];
    d2 += o.x * rk[64 + w] + o.y * rk[65 + w] + o.z * rk[66 + w] + o.w * rk[67 + w];
    d3 += o.x * rk[96 + w] + o.y * rk[97 + w] + o.z * rk[98 + w] + o.w * rk[99 + w];
  }
  float nrm = sqrtf(nn);
  rsim[(b * R_ + 0) * N_ + n] = d0 / (nrm * rkn[0] + EPS_);
  rsim[(b * R_ + 1) * N_ + n] = d1 / (nrm * rkn[1] + EPS_);
  rsim[(b * R_ + 2) * N_ + n] = d2 / (nrm * rkn[2] + EPS_);
  rsim[(b * R_ + 3) * N_ + n] = d3 / (nrm * rkn[3] + EPS_);
}

// ---------------- zero scratch accumulators ----------------
__global__ void zero_kernel(float* __restrict__ p, int n) {
  int t = blockIdx.x * blockDim.x + threadIdx.x;
  if (t < n) p[t] = 0.0f;
}

// ---------------- K6: fused link update + fw (L·w) + bw (L^T·w) ----------------
// Streams the 134MB link tensor exactly once (b128 load + b128 store), with ZERO
// barriers in the hot loop: bw partials accumulate via LDS float atomics
// (ds_add_f32) into s_bw[4][512]; fw accumulates in registers. One barrier at the
// end, then LDS-tree fw reduce (block owns its n-strip) and a cooperative global
// atomic flush of the bw tile (32 n-strip blocks contribute per (b,r,m)).
__global__ __launch_bounds__(256) void link_kernel(const float* __restrict__ link_in,
                                                   const float* __restrict__ prec_old,
                                                   const float* __restrict__ rw_old,
                                                   const float* __restrict__ ww,
                                                   float* __restrict__ link_out,
                                                   float* __restrict__ fw,
                                                   float* __restrict__ bw) {
  __shared__ float s_rw[R_ * 512];   // read_weights rows for this batch (8KB)
  __shared__ float s_ww[512];
  __shared__ float s_pr[512];
  __shared__ float s_bw[R_ * 512];   // LDS bw accumulators (8KB)
  __shared__ float red[256];
  int b = blockIdx.y, n0 = blockIdx.x * 16, t = threadIdx.x;
  int tn = t >> 4, tm = t & 15;
  for (int i = t; i < R_ * 512; i += 256) { s_rw[i] = rw_old[(size_t)b * R_ * N_ + i]; s_bw[i] = 0.0f; }
  for (int i = t; i < 512; i += 256) { s_ww[i] = ww[b * N_ + i]; s_pr[i] = prec_old[b * N_ + i]; }
  __syncthreads();
  int n = n0 + tn;
  float wwn = s_ww[n];
  float rwn0 = s_rw[n], rwn1 = s_rw[512 + n], rwn2 = s_rw[1024 + n], rwn3 = s_rw[1536 + n];
  float f0 = 0.f, f1 = 0.f, f2 = 0.f, f3 = 0.f;
  const float* lin  = link_in  + (size_t)b * N_ * N_ + (size_t)n * N_;
  float*       lout = link_out + (size_t)b * N_ * N_ + (size_t)n * N_;
  for (int m0 = 0; m0 < N_; m0 += 64) {          // 8 iterations, 64 m per chunk
    int m = m0 + tm * 4;
    __builtin_prefetch(lin + m + 512, 0, 1);     // global_prefetch ahead of the stream
    float4 lv = *(const float4*)(lin + m);
    float4 ln;
    ln.x = (1.0f - wwn - s_ww[m + 0]) * lv.x + wwn * s_pr[m + 0];
    ln.y = (1.0f - wwn - s_ww[m + 1]) * lv.y + wwn * s_pr[m + 1];
    ln.z = (1.0f - wwn - s_ww[m + 2]) * lv.z + wwn * s_pr[m + 2];
    ln.w = (1.0f - wwn - s_ww[m + 3]) * lv.w + wwn * s_pr[m + 3];
    if (n == m + 0) ln.x = 0.0f;
    if (n == m + 1) ln.y = 0.0f;
    if (n == m + 2) ln.z = 0.0f;
    if (n == m + 3) ln.w = 0.0f;
    *(float4*)(lout + m) = ln;
    // fw: register accumulation against the 4 rw rows
    float4 r0 = *(const float4*)&s_rw[m];
    float4 r1 = *(const float4*)&s_rw[512 + m];
    float4 r2 = *(const float4*)&s_rw[1024 + m];
    float4 r3 = *(const float4*)&s_rw[1536 + m];
    f0 += ln.x * r0.x + ln.y * r0.y + ln.z * r0.z + ln.w * r0.w;
    f1 += ln.x * r1.x + ln.y * r1.y + ln.z * r1.z + ln.w * r1.w;
    f2 += ln.x * r2.x + ln.y * r2.y + ln.z * r2.z + ln.w * r2.w;
    f3 += ln.x * r3.x + ln.y * r3.y + ln.z * r3.z + ln.w * r3.w;
    // bw partials: LDS float atomics, no barriers (ds_add_f32)
    atomicAdd(&s_bw[m + 0], ln.x * rwn0); atomicAdd(&s_bw[m + 1], ln.y * rwn0);
    atomicAdd(&s_bw[m + 2], ln.z * rwn0); atomicAdd(&s_bw[m + 3], ln.w * rwn0);
    atomicAdd(&s_bw[512 + m + 0], ln.x * rwn1); atomicAdd(&s_bw[512 + m + 1], ln.y * rwn1);
    atomicAdd(&s_bw[512 + m + 2], ln.z * rwn1); atomicAdd(&s_bw[512 + m + 3], ln.w * rwn1);
    atomicAdd(&s_bw[1024 + m + 0], ln.x * rwn2); atomicAdd(&s_bw[1024 + m + 1], ln.y * rwn2);
    atomicAdd(&s_bw[1024 + m + 2], ln.z * rwn2); atomicAdd(&s_bw[1024 + m + 3], ln.w * rwn2);
    atomicAdd(&s_bw[1536 + m + 0], ln.x * rwn3); atomicAdd(&s_bw[1536 + m + 1], ln.y * rwn3);
    atomicAdd(&s_bw[1536 + m + 2], ln.z * rwn3); atomicAdd(&s_bw[1536 + m + 3], ln.w * rwn3);
  }
  __syncthreads();
  // flush bw tile to global (32 n-strip blocks contribute per (b,r,m))
  for (int i = t; i < R_ * 512; i += 256)
    atomicAdd(&bw[(size_t)b * R_ * N_ + i], s_bw[i]);
  // fw: deterministic tree reduce over tm, direct store (block owns n-strip)
  float fv[4] = { f0, f1, f2, f3 };
  for (int r = 0; r < R_; ++r) {
    red[t] = fv[r]; __syncthreads();
    for (int s = 8; s > 0; s >>= 1) { if (tm < s) red[t] += red[t + s]; __syncthreads(); }
    if (tm == 0) fw[((size_t)b * R_ + r) * N_ + n] = red[t];
    __syncthreads();
  }
}

// ---------------- K7: read softmax + mode mixing + read vectors ----------------
__global__ __launch_bounds__(256) void read_kernel(const float* __restrict__ ifc,
                                                   const float* __restrict__ rsim,
                                                   const float* __restrict__ fw,
                                                   const float* __restrict__ bw,
                                                   const float* __restrict__ mem_new,
                                                   float* __restrict__ rw_out,
                                                   float* __restrict__ rv_out) {
  __shared__ float s_rw[512];
  __shared__ float red[256];
  int br = blockIdx.x;           // b*R + r
  int b = br >> 2, r = br & 3;
  int t = threadIdx.x;
  float rstr = ifc[b * C_ + 128 + r];
  float m0v = ifc[b * C_ + 235 + r * 3 + 0];
  float m1v = ifc[b * C_ + 235 + r * 3 + 1];
  float m2v = ifc[b * C_ + 235 + r * 3 + 2];
  float mm = fmaxf(m0v, fmaxf(m1v, m2v));
  float e0 = __expf(m0v - mm), e1 = __expf(m1v - mm), e2 = __expf(m2v - mm);
  float es = e0 + e1 + e2; e0 /= es; e1 /= es; e2 /= es;
  const float* rs = rsim + (size_t)br * N_;
  float sc0 = rstr * rs[t], sc1 = rstr * rs[t + 256];
  red[t] = fmaxf(sc0, sc1); __syncthreads();
  for (int s = 128; s > 0; s >>= 1) { if (t < s) red[t] = fmaxf(red[t], red[t + s]); __syncthreads(); }
  float mx = red[0]; __syncthreads();
  float ex0 = __expf(sc0 - mx), ex1 = __expf(sc1 - mx);
  red[t] = ex0 + ex1; __syncthreads();
  for (int s = 128; s > 0; s >>= 1) { if (t < s) red[t] += red[t + s]; __syncthreads(); }
  float inv = 1.0f / red[0]; __syncthreads();
  const float* fwp = fw + (size_t)br * N_;
  const float* bwp = bw + (size_t)br * N_;
  float w0 = e0 * bwp[t]       + e1 * ex0 * inv + e2 * fwp[t];
  float w1 = e0 * bwp[t + 256] + e1 * ex1 * inv + e2 * fwp[t + 256];
  rw_out[(size_t)br * N_ + t] = w0;
  rw_out[(size_t)br * N_ + t + 256] = w1;
  s_rw[t] = w0; s_rw[t + 256] = w1;
  __syncthreads();
  int w = t & 31, chunk = t >> 5;                // 8 chunks x 64 rows each
  const float* mb = mem_new + (size_t)b * N_ * W_;
  float acc = 0.f;
  for (int n = chunk * 64; n < chunk * 64 + 64; ++n) acc += s_rw[n] * mb[(size_t)n * W_ + w];
  red[t] = acc; __syncthreads();
  for (int s = 4; s > 0; s >>= 1) { if (chunk < s) red[t] += red[t + (s << 5)]; __syncthreads(); }
  if (chunk == 0) rv_out[(size_t)br * W_ + w] = red[t];
}

// ---------------- launcher ----------------
extern "C" void kernel_launch(void* const* d_in, const int* in_sizes, int n_in,
                              void* d_out, int out_size, void* d_ws, size_t ws_size,
                              hipStream_t stream) {
  const float* xi       = (const float*)d_in[0];
  const float* mem_in   = (const float*)d_in[1];
  const float* link_in  = (const float*)d_in[2];
  const float* prec_in  = (const float*)d_in[3];
  const float* rw_in    = (const float*)d_in[4];
  const float* wwgt_in  = (const float*)d_in[5];
  const float* usage_in = (const float*)d_in[6];
  const float* Wp[10], *bp[10];
  for (int i = 0; i < 10; ++i) { Wp[i] = (const float*)d_in[7 + 2 * i]; bp[i] = (const float*)d_in[8 + 2 * i]; }

  float* out = (float*)d_out;
  float* rv_o   = out + OUT_RV;
  float* mem_o  = out + OUT_MEM;
  float* link_o = out + OUT_LINK;
  float* prec_o = out + OUT_PREC;
  float* rw_o   = out + OUT_RW;
  float* ww_o   = out + OUT_WW;
  float* use_o  = out + OUT_USAGE;

  float* ws    = (float*)d_ws;
  float* Wcat  = ws + WS_WCAT;
  float* bcat  = ws + WS_BCAT;
  float* ifc   = ws + WS_IFC;
  float* alloc = ws + WS_ALLOC;
  float* sww   = ws + WS_SWW;
  float* rsim  = ws + WS_RSIM;
  float* fwbuf = ws + WS_FW;
  float* bwbuf = ws + WS_BW;

  // K0: pack head weights
  pack_kernel<<<(C_ * I_ + 255) / 256, 256, 0, stream>>>(
      Wp[0], Wp[1], Wp[2], Wp[3], Wp[4], Wp[5], Wp[6], Wp[7], Wp[8], Wp[9],
      bp[0], bp[1], bp[2], bp[3], bp[4], bp[5], bp[6], bp[7], bp[8], bp[9],
      Wcat, bcat);

  // K1: WMMA heads GEMM + activations
  heads_gemm_kernel<<<dim3(B_ / 16, C_ / 16), 32, 0, stream>>>(xi, Wcat, bcat, ifc);

  // K2: usage update
  usage_kernel<<<(B_ * N_ + 255) / 256, 256, 0, stream>>>(ifc, rw_in, wwgt_in, usage_in, use_o);

  // K3: allocation (per-batch bitonic sort)
  alloc_kernel<<<B_, 512, 0, stream>>>(use_o, alloc);

  // K4: write content addressing + ww + precedence
  ww_kernel<<<B_, 512, 0, stream>>>(ifc, mem_in, alloc, prec_in, ww_o, prec_o, sww);

  // K5: memory write + read-content similarities
  memwrite_kernel<<<dim3(N_ / 256, B_), 256, 0, stream>>>(ifc, mem_in, ww_o, mem_o, rsim);

  // zero fw/bw accumulators (deterministic re-runs)
  zero_kernel<<<(2 * B_ * R_ * N_ + 255) / 256, 256, 0, stream>>>(fwbuf, 2 * B_ * R_ * N_);

  // K6: fused link update + fw + bw (single barrier-free pass over 134MB link tensor)
  link_kernel<<<dim3(N_ / 16, B_), 256, 0, stream>>>(link_in, prec_in, rw_in, ww_o,
                                                     link_o, fwbuf, bwbuf);

  // K7: read weights + read vectors
  read_kernel<<<B_ * R_, 256, 0, stream>>>(ifc, rsim, fwbuf, bwbuf, mem_o, rw_o, rv_o);
}